// EMGNN_37924561224001
// MI455X (gfx1250) — compile-verified
//
#include <hip/hip_runtime.h>

#define LEAKY_ALPHA 0.2f

typedef float v2f __attribute__((ext_vector_type(2)));
typedef float v8f __attribute__((ext_vector_type(8)));

// ---------------------------------------------------------------------------
// C[M x Ncols] = A[M x 128] * W[128 x Ncols] (+ bias) (+ leaky relu)
//
// Block = 64-row strip. A (64x128) and W^T (ncolsPad x 128) staged in LDS with
// a 132-float pitch (bank step 4 -> conflict free). Each wave owns one 16-col
// tile and iterates 4 row tiles, so one B fragment feeds 4 WMMAs.
// Native fp32 WMMA: V_WMMA_F32_16X16X4_F32, K stepped by 4.
//
// 16x16x4 F32 layouts (ISA 7.12.2):
//   A (16x4):  lanes 0-15 row=lane  {K=k,K=k+1}; lanes 16-31 row=lane-16 {K=k+2,K=k+3}
//   B (4x16):  lanes 0-15 col=lane  {K=k,K=k+1}; lanes 16-31 col=lane-16 {K=k+2,K=k+3}
//   C/D (v8f): VGPR r -> row = r + 8*(lane>=16), col = lane&15
// ---------------------------------------------------------------------------
template <int NCOLT>   // number of 16-wide column tiles (8 for H=128, 3 for 40)
__global__ void gemm_k128_wmma(const float* __restrict__ A,
                               const float* __restrict__ W,
                               const float* __restrict__ bias,
                               float* __restrict__ C,
                               int M, int Ncols, int applyLeaky)
{
    constexpr int NCP = NCOLT * 16;          // padded column count
    __shared__ float lds_a[64 * 132];        // A strip, row-major, pitch 132
    __shared__ float lds_w[NCP * 132];       // W transposed: [col][k], pitch 132

    const int rowBase = blockIdx.x * 64;

    // stage A (64 x 128), zero-pad OOB rows
    for (int idx = threadIdx.x; idx < 64 * 128; idx += blockDim.x) {
        const int r = idx >> 7, c = idx & 127;
        const int gr = rowBase + r;
        lds_a[r * 132 + c] = (gr < M) ? A[(size_t)gr * 128 + c] : 0.0f;
    }
    // stage W transposed (coalesced global read), zero-pad cols >= Ncols
    for (int idx = threadIdx.x; idx < 128 * NCP; idx += blockDim.x) {
        const int k = idx / NCP;             // NCP is compile-time
        const int c = idx - k * NCP;
        lds_w[c * 132 + k] = (c < Ncols) ? W[k * Ncols + c] : 0.0f;
    }
    __syncthreads();

    const int wave = threadIdx.x >> 5;
    const int lane = threadIdx.x & 31;
    if (wave >= NCOLT) return;

    const int half = lane >> 4;              // 0: lanes 0-15, 1: lanes 16-31
    const int l15  = lane & 15;
    const int col  = (wave << 4) + l15;

    v8f acc[4];
    #pragma unroll
    for (int t = 0; t < 4; ++t) acc[t] = v8f{0.f,0.f,0.f,0.f,0.f,0.f,0.f,0.f};

    const float* wp = lds_w + col * 132 + half * 2;     // b.x/b.y contiguous
    const float* ap = lds_a + l15 * 132 + half * 2;

    #pragma unroll 4
    for (int k = 0; k < 128; k += 4) {
        const v2f b = *(const v2f*)(wp + k);
        #pragma unroll
        for (int t = 0; t < 4; ++t) {
            const v2f a = *(const v2f*)(ap + t * (16 * 132) + k);
            // (neg_a, A, neg_b, B, c_mod, C, reuse_a, reuse_b)
            acc[t] = __builtin_amdgcn_wmma_f32_16x16x4_f32(false, a, false, b,
                                                           (short)0, acc[t],
                                                           false, false);
        }
    }

    const bool colOK = (col < Ncols);
    const float bv = (bias != nullptr && colOK) ? bias[col] : 0.0f;
    #pragma unroll
    for (int t = 0; t < 4; ++t) {
        #pragma unroll
        for (int r = 0; r < 8; ++r) {
            const int row = rowBase + t * 16 + r + half * 8;
            if (row < M && colOK) {
                float v = acc[t][r] + bv;
                if (applyLeaky) v = (v > 0.0f) ? v : LEAKY_ALPHA * v;
                C[(size_t)row * Ncols + col] = v;
            }
        }
    }
}

// ---------------------------------------------------------------------------
// Degree / normalization helpers (self-loop contributes deg init of 1)
// ---------------------------------------------------------------------------
__global__ void deg_init(float* __restrict__ deg, int n)
{
    const int i = blockIdx.x * blockDim.x + threadIdx.x;
    if (i < n) deg[i] = 1.0f;   // self-loop
}

__global__ void deg_count(const int* __restrict__ dst, int nE, float* __restrict__ deg)
{
    const int e = blockIdx.x * blockDim.x + threadIdx.x;
    if (e < nE) atomicAdd(&deg[dst[e]], 1.0f);
}

__global__ void deg_to_dinv(const float* __restrict__ deg, float* __restrict__ dinv, int n)
{
    const int i = blockIdx.x * blockDim.x + threadIdx.x;
    if (i < n) dinv[i] = rsqrtf(deg[i]);   // deg >= 1 always (self-loop)
}

// out[v,:] = hW[v,:] * dinv[v]^2   (the self-loop message, also initializes out)
__global__ void self_loop_init(const float* __restrict__ hW,
                               const float* __restrict__ dinv,
                               float* __restrict__ out, int n)
{
    const int i = blockIdx.x * blockDim.x + threadIdx.x;
    if (i < n * 128) {
        const float di = dinv[i >> 7];
        out[i] = hW[i] * di * di;
    }
}

// one wave per edge: out[dst,:] += hW[src,:] * dinv[src]*dinv[dst]
__global__ void scatter_add_edges(const float* __restrict__ hW,
                                  const int* __restrict__ src,
                                  const int* __restrict__ dst,
                                  const float* __restrict__ dinv,
                                  float* __restrict__ out, int nE)
{
    const int e = blockIdx.x * (blockDim.x >> 5) + (threadIdx.x >> 5);
    if (e >= nE) return;
    const int lane = threadIdx.x & 31;
    const int s = src[e], d = dst[e];
    const float norm = dinv[s] * dinv[d];
    const float4 v = ((const float4*)(hW + (size_t)s * 128))[lane];
    float* op = out + (size_t)d * 128 + lane * 4;
    atomicAdd(op + 0, v.x * norm);
    atomicAdd(op + 1, v.y * norm);
    atomicAdd(op + 2, v.z * norm);
    atomicAdd(op + 3, v.w * norm);
}

// h = leaky(agg + bias)
__global__ void bias_leaky(const float* __restrict__ in, const float* __restrict__ b,
                           float* __restrict__ h, int n)
{
    const int i = blockIdx.x * blockDim.x + threadIdx.x;
    if (i < n * 128) {
        const float v = in[i] + b[i & 127];
        h[i] = (v > 0.0f) ? v : LEAKY_ALPHA * v;
    }
}

// in-place row-wise log_softmax over 40 columns; one wave32 per row
__global__ void log_softmax40(float* __restrict__ logits, int rows)
{
    const int row = blockIdx.x * (blockDim.x >> 5) + (threadIdx.x >> 5);
    if (row >= rows) return;
    const int lane = threadIdx.x & 31;
    float* p = logits + (size_t)row * 40;

    const float v0 = p[lane];
    const float v1 = (lane < 8) ? p[32 + lane] : -3.402823466e38f;

    float m = fmaxf(v0, v1);
    for (int o = 16; o > 0; o >>= 1) m = fmaxf(m, __shfl_xor(m, o, 32));

    float s = expf(v0 - m) + ((lane < 8) ? expf(v1 - m) : 0.0f);
    for (int o = 16; o > 0; o >>= 1) s += __shfl_xor(s, o, 32);

    const float ls = logf(s);
    p[lane] = v0 - m - ls;
    if (lane < 8) p[32 + lane] = v1 - m - ls;
}

// ---------------------------------------------------------------------------
extern "C" void kernel_launch(void* const* d_in, const int* in_sizes, int n_in,
                              void* d_out, int out_size, void* d_ws, size_t ws_size,
                              hipStream_t stream)
{
    const float* x      = (const float*)d_in[0];
    const float* meta_x = (const float*)d_in[1];
    const int*   ei     = (const int*)d_in[2];
    const int*   mei    = (const int*)d_in[3];
    const float* lin_W  = (const float*)d_in[4];
    const float* lin_b  = (const float*)d_in[5];
    const float* mlin_W = (const float*)d_in[6];
    const float* mlin_b = (const float*)d_in[7];
    const float* conv_W = (const float*)d_in[8];
    const float* conv_b = (const float*)d_in[9];
    const float* mg_W   = (const float*)d_in[10];
    const float* mg_b   = (const float*)d_in[11];
    const float* clf_W  = (const float*)d_in[12];
    const float* clf_b  = (const float*)d_in[13];

    const int N  = in_sizes[0] / 128;   // 50000
    const int Mm = in_sizes[1] / 128;   // 500
    const int NT = N + Mm;              // 50500
    const int E  = in_sizes[2] / 2;     // 800000
    const int ME = in_sizes[3] / 2;     // 50000
    const int H  = 128;

    char* ws = (char*)d_ws;
    size_t off = 0;
    auto carve = [&](size_t bytes) -> void* {
        off = (off + 255) & ~(size_t)255;
        void* p = ws + off;
        off += bytes;
        return p;
    };
    float* hcat  = (float*)carve((size_t)NT * H * 4);
    float* tmp   = (float*)carve((size_t)NT * H * 4);
    float* agg   = (float*)carve((size_t)NT * H * 4);
    float* deg   = (float*)carve((size_t)NT * 4);
    float* dinvN = (float*)carve((size_t)N  * 4);
    float* dinvT = (float*)carve((size_t)NT * 4);
    (void)ws_size; (void)n_in;

    // 1) h = leaky(x @ lin_W + lin_b)  -> hcat[0:N]
    gemm_k128_wmma<8><<<(N + 63) / 64, 256, 0, stream>>>(x, lin_W, lin_b, hcat, N, H, 1);
    // 2) mh = leaky(meta_x @ meta_lin_W + meta_lin_b) -> hcat[N:NT]
    gemm_k128_wmma<8><<<(Mm + 63) / 64, 256, 0, stream>>>(meta_x, mlin_W, mlin_b,
                                                          hcat + (size_t)N * H, Mm, H, 1);

    // 3) node-graph degrees -> dinvN
    deg_init<<<(N + 255) / 256, 256, 0, stream>>>(deg, N);
    deg_count<<<(E + 255) / 256, 256, 0, stream>>>(ei + E, E, deg);
    deg_to_dinv<<<(N + 255) / 256, 256, 0, stream>>>(deg, dinvN, N);

    // 4) two GCN layers on hcat[0:N]
    for (int i = 0; i < 2; ++i) {
        gemm_k128_wmma<8><<<(N + 63) / 64, 256, 0, stream>>>(
            hcat, conv_W + (size_t)i * H * H, nullptr, tmp, N, H, 0);
        self_loop_init<<<((N * H) + 255) / 256, 256, 0, stream>>>(tmp, dinvN, agg, N);
        scatter_add_edges<<<(E + 7) / 8, 256, 0, stream>>>(tmp, ei, ei + E, dinvN, agg, E);
        bias_leaky<<<((N * H) + 255) / 256, 256, 0, stream>>>(agg, conv_b + (size_t)i * H,
                                                              hcat, N);
    }

    // 5) meta-graph degrees over NT nodes -> dinvT
    deg_init<<<(NT + 255) / 256, 256, 0, stream>>>(deg, NT);
    deg_count<<<(ME + 255) / 256, 256, 0, stream>>>(mei + ME, ME, deg);
    deg_to_dinv<<<(NT + 255) / 256, 256, 0, stream>>>(deg, dinvT, NT);

    // 6) meta GCN over the concatenated matrix
    gemm_k128_wmma<8><<<(NT + 63) / 64, 256, 0, stream>>>(hcat, mg_W, nullptr, tmp, NT, H, 0);
    self_loop_init<<<((NT * H) + 255) / 256, 256, 0, stream>>>(tmp, dinvT, agg, NT);
    scatter_add_edges<<<(ME + 7) / 8, 256, 0, stream>>>(tmp, mei, mei + ME, dinvT, agg, ME);
    bias_leaky<<<((NT * H) + 255) / 256, 256, 0, stream>>>(agg, mg_b, hcat, NT);

    // 7) classifier: logits = hcat @ clf_W + clf_b  (40 cols -> 3 col tiles)
    gemm_k128_wmma<3><<<(NT + 63) / 64, 256, 0, stream>>>(hcat, clf_W, clf_b,
                                                          (float*)d_out, NT, 40, 0);

    // 8) in-place row log_softmax
    log_softmax40<<<(NT + 7) / 8, 256, 0, stream>>>((float*)d_out, NT);
    (void)out_size;
}